// GNN_Model_2078764171910
// MI455X (gfx1250) — compile-verified
//
#include <hip/hip_runtime.h>
#include <hip/hip_bf16.h>
#include <math.h>

typedef float v2f __attribute__((ext_vector_type(2)));
typedef float v8f __attribute__((ext_vector_type(8)));

#define N_P   50000
#define L_HOP 5
#define N_LNK 20000
#define N_QUE 60000
#define QPL   3
#define DIM   64
#define RU    256
#define NPOL  3
#define MAXQ  3
#define T_IT  4
#define GDIM  192          // 3*DIM
#define N_E   (N_P * L_HOP)

// ---------------- WMMA helpers (f32 16x16x4, full precision) ----------------

__device__ __forceinline__ v8f wmma4(v2f a, v2f b, v8f c) {
  // 8 args: (neg_a, A, neg_b, B, c_mod, C, reuse_a, reuse_b)
  return __builtin_amdgcn_wmma_f32_16x16x4_f32(false, a, false, b, (short)0, c,
                                               false, false);
}

// Dual-accumulator GEMM: C0 += A*B[:,n0..], C1 += A*B[:,n1..]
// A: 16 x 4*NKT in LDS (row stride sa floats)
// Bp: paired layout, Bp[p*N+n] = float2{B[2p][n], B[2p+1][n]}
// lane roles: m = lane&15 (A-row == B/C-col), kh = lane>>4 (K-half / M-half)
// K-loop kept rolled so B loads cannot be hoisted & spilled to scratch.
template <int NKT>
__device__ __forceinline__ void gemm_dual(const float* __restrict__ A, int sa,
                                          const float* __restrict__ Bp, int N,
                                          int n0, int n1, int m, int kh,
                                          v8f& c0, v8f& c1) {
  const float* ar = A + m * sa + kh * 2;
  const float* b0 = Bp + ((size_t)kh * N + (n0 + m)) * 2;
  const float* b1 = Bp + ((size_t)kh * N + (n1 + m)) * 2;
  const size_t bstep = (size_t)4 * N;  // 2 pair-rows per kt, 2 floats per pair
#pragma unroll 2
  for (int kt = 0; kt < NKT; ++kt) {
    v2f a = *(const v2f*)(ar + kt * 4);
    v2f bb0 = *(const v2f*)(b0 + kt * bstep);
    v2f bb1 = *(const v2f*)(b1 + kt * bstep);
    c0 = wmma4(a, bb0, c0);
    c1 = wmma4(a, bb1, c1);
  }
}

template <int NKT>
__device__ __forceinline__ void gemm_one(const float* __restrict__ A, int sa,
                                         const float* __restrict__ Bp, int N,
                                         int n0, int m, int kh, v8f& c0) {
  const float* ar = A + m * sa + kh * 2;
  const float* b0 = Bp + ((size_t)kh * N + (n0 + m)) * 2;
  const size_t bstep = (size_t)4 * N;
#pragma unroll 2
  for (int kt = 0; kt < NKT; ++kt) {
    v2f a = *(const v2f*)(ar + kt * 4);
    v2f bb0 = *(const v2f*)(b0 + kt * bstep);
    c0 = wmma4(a, bb0, c0);
  }
}

__device__ __forceinline__ float sigm(float x) { return 1.f / (1.f + expf(-x)); }

__device__ __forceinline__ float selu_f(float x) {
  const float sc = 1.0507009873554805f, al = 1.6732632423543772f;
  return x > 0.f ? sc * x : sc * al * (expf(x) - 1.f);
}

// one Keras GRU (reset_after=True) step entirely on WMMA.
// X: [16 x 4*NKX] input in LDS (stride sx), H: [16 x 64] hidden in LDS (stride sh)
// kP: paired kernel [(4*NKX)/2][192], rkP: paired recurrent [32][192]
// b: biases, b[0..191]=input bias, b[192..383]=recurrent bias
// hreg: hidden in C-register layout, updated in place (LDS H left untouched)
template <int NKX>
__device__ __forceinline__ void gru_step(const float* __restrict__ X, int sx,
                                         const float* __restrict__ H, int sh,
                                         const float* __restrict__ kP,
                                         const float* __restrict__ rkP,
                                         const float* __restrict__ b,
                                         int m, int kh, v8f hreg[4]) {
#pragma unroll 1
  for (int j = 0; j < 4; ++j) {
    // z gate (cols j*16..) and r gate (cols 64+j*16..) share A operands
    v8f zc = {}, rc = {};
    gemm_dual<NKX>(X, sx, kP, GDIM, j * 16, 64 + j * 16, m, kh, zc, rc);
    gemm_dual<16>(H, sh, rkP, GDIM, j * 16, 64 + j * 16, m, kh, zc, rc);
    // candidate: x-part and h-part kept separate (reset gate applies to h-part)
    v8f xc = {}, hc = {};
    gemm_one<NKX>(X, sx, kP, GDIM, 128 + j * 16, m, kh, xc);
    gemm_one<16>(H, sh, rkP, GDIM, 128 + j * 16, m, kh, hc);

    const int col = j * 16 + m;
    const float bz = b[col] + b[GDIM + col];
    const float br = b[64 + col] + b[GDIM + 64 + col];
    const float bx = b[128 + col];
    const float bh = b[GDIM + 128 + col];
#pragma unroll
    for (int v = 0; v < 8; ++v) {
      float z = sigm(zc[v] + bz);
      float r = sigm(rc[v] + br);
      float cand = tanhf(xc[v] + bx + r * (hc[v] + bh));
      hreg[j][v] = z * hreg[j][v] + (1.f - z) * cand;
    }
  }
}

// ---------------- init / prep kernels ----------------

__global__ void init_path_k(const float* __restrict__ traffic,
                            const float* __restrict__ packets,
                            float* __restrict__ ps) {
  int tid = blockIdx.x * 256 + threadIdx.x;
  if (tid >= N_P * DIM) return;
  int p = tid >> 6, d = tid & 63;
  ps[tid] = (d == 0) ? traffic[p] : (d == 1) ? packets[p] : 0.f;
}

__global__ void init_link_k(const float* __restrict__ cap,
                            const int* __restrict__ policy,
                            float* __restrict__ ls) {
  int tid = blockIdx.x * 256 + threadIdx.x;
  if (tid >= N_LNK * DIM) return;
  int l = tid >> 6, d = tid & 63;
  float v = 0.f;
  if (d == 0) v = cap[l];
  else if (d >= 1 && d <= NPOL && policy[l] == d - 1) v = 1.f;
  ls[tid] = v;
}

__global__ void init_queue_k(const float* __restrict__ qsize,
                             const float* __restrict__ weight,
                             const int* __restrict__ prio,
                             float* __restrict__ qs) {
  int tid = blockIdx.x * 256 + threadIdx.x;
  if (tid >= N_QUE * DIM) return;
  int q = tid >> 6, d = tid & 63;
  float v = 0.f;
  if (d == 0) v = qsize[q];
  else if (d >= 1 && d <= MAXQ && prio[q] == d - 1) v = 1.f;
  else if (d == MAXQ + 1) v = weight[q];
  qs[tid] = v;
}

// pack B[K][N] (row major) into pair layout Bp[K/2][N] of float2
__global__ void pair_weights_k(const float* __restrict__ B,
                               float* __restrict__ Bp, int K, int N) {
  int tid = blockIdx.x * 256 + threadIdx.x;
  int tot = (K / 2) * N;
  if (tid >= tot) return;
  int p = tid / N, n = tid - p * N;
  Bp[2 * (size_t)tid + 0] = B[(size_t)(2 * p) * N + n];
  Bp[2 * (size_t)tid + 1] = B[(size_t)(2 * p + 1) * N + n];
}

__global__ void zero_k(float* __restrict__ p, int n) {
  int tid = blockIdx.x * 256 + threadIdx.x;
  if (tid < n) p[tid] = 0.f;
}

__global__ void scatter_add_k(const float* __restrict__ ps,
                              const int* __restrict__ p2q,
                              const int* __restrict__ seqq,
                              float* __restrict__ psum) {
  int tid = blockIdx.x * 256 + threadIdx.x;
  if (tid >= N_E * DIM) return;
  int e = tid >> 6, d = tid & 63;
  float val = ps[(size_t)p2q[e] * DIM + d];
  __hip_atomic_fetch_add(&psum[(size_t)seqq[e] * DIM + d], val,
                         __ATOMIC_RELAXED, __HIP_MEMORY_SCOPE_AGENT);
}

// ---------------- GRU kernels ----------------

// 16 paths / wave; 5-step GRU with per-hop gathers of queue & link state
__global__ __launch_bounds__(32)
void path_update_k(float* __restrict__ ps, const float* __restrict__ qs,
                   const float* __restrict__ ls, const float* __restrict__ pkP,
                   const float* __restrict__ prkP, const float* __restrict__ pb,
                   const int* __restrict__ l2p, const int* __restrict__ q2p) {
  __shared__ float X[16 * 132];  // [16 x 128] inputs, padded stride
  __shared__ float H[16 * 68];   // [16 x 64] hidden
  const int lane = threadIdx.x;
  const int m = lane & 15, kh = lane >> 4;
  const int p0 = blockIdx.x * 16;

  v8f hreg[4];
#pragma unroll
  for (int j = 0; j < 4; ++j)
#pragma unroll
    for (int v = 0; v < 8; ++v) {
      float val = ps[(size_t)(p0 + v + 8 * kh) * DIM + j * 16 + m];
      hreg[j][v] = val;
      H[(v + 8 * kh) * 68 + j * 16 + m] = val;
    }
  __syncthreads();

  const int r = lane >> 1, half = lane & 1;
#pragma unroll 1
  for (int t = 0; t < L_HOP; ++t) {
    // gather x = concat(queue_state[q], link_state[l]) for hop t
    int e = (p0 + r) * L_HOP + t;
    const float* src = half ? (ls + (size_t)l2p[e] * DIM)
                            : (qs + (size_t)q2p[e] * DIM);
    float* dst = X + r * 132 + half * 64;
#pragma unroll
    for (int c = 0; c < 64; c += 4)
      *(float4*)(dst + c) = *(const float4*)(src + c);
    __syncthreads();

    gru_step<32>(X, 132, H, 68, pkP, prkP, pb, m, kh, hreg);
    __syncthreads();
#pragma unroll
    for (int j = 0; j < 4; ++j)
#pragma unroll
      for (int v = 0; v < 8; ++v)
        H[(v + 8 * kh) * 68 + j * 16 + m] = hreg[j][v];
    __syncthreads();
  }
#pragma unroll
  for (int j = 0; j < 4; ++j)
#pragma unroll
    for (int v = 0; v < 8; ++v)
      ps[(size_t)(p0 + v + 8 * kh) * DIM + j * 16 + m] = hreg[j][v];
}

// 16 queues / wave; single GRU cell with x = path_sum
__global__ __launch_bounds__(32)
void queue_update_k(float* __restrict__ qs, const float* __restrict__ psum,
                    const float* __restrict__ qkP,
                    const float* __restrict__ qrkP,
                    const float* __restrict__ qb) {
  __shared__ float X[16 * 68];
  __shared__ float H[16 * 68];
  const int lane = threadIdx.x;
  const int m = lane & 15, kh = lane >> 4;
  const int q0 = blockIdx.x * 16;

  v8f hreg[4];
#pragma unroll
  for (int j = 0; j < 4; ++j)
#pragma unroll
    for (int v = 0; v < 8; ++v) {
      float val = qs[(size_t)(q0 + v + 8 * kh) * DIM + j * 16 + m];
      hreg[j][v] = val;
      H[(v + 8 * kh) * 68 + j * 16 + m] = val;
    }
  const int r = lane >> 1, half = lane & 1;
  {
    const float* src = psum + (size_t)(q0 + r) * DIM + half * 32;
    float* dst = X + r * 68 + half * 32;
#pragma unroll
    for (int c = 0; c < 32; c += 4)
      *(float4*)(dst + c) = *(const float4*)(src + c);
  }
  __syncthreads();

  gru_step<16>(X, 68, H, 68, qkP, qrkP, qb, m, kh, hreg);

#pragma unroll
  for (int j = 0; j < 4; ++j)
#pragma unroll
    for (int v = 0; v < 8; ++v)
      qs[(size_t)(q0 + v + 8 * kh) * DIM + j * 16 + m] = hreg[j][v];
}

// 16 links / wave; 3-step GRU over the link's queues (q = link*3 + t)
__global__ __launch_bounds__(32)
void link_update_k(float* __restrict__ ls, const float* __restrict__ qs,
                   const float* __restrict__ lkP,
                   const float* __restrict__ lrkP,
                   const float* __restrict__ lb) {
  __shared__ float X[16 * 68];
  __shared__ float H[16 * 68];
  const int lane = threadIdx.x;
  const int m = lane & 15, kh = lane >> 4;
  const int l0 = blockIdx.x * 16;

  v8f hreg[4];
#pragma unroll
  for (int j = 0; j < 4; ++j)
#pragma unroll
    for (int v = 0; v < 8; ++v) {
      float val = ls[(size_t)(l0 + v + 8 * kh) * DIM + j * 16 + m];
      hreg[j][v] = val;
      H[(v + 8 * kh) * 68 + j * 16 + m] = val;
    }
  __syncthreads();

  const int r = lane >> 1, half = lane & 1;
#pragma unroll 1
  for (int t = 0; t < QPL; ++t) {
    int q = (l0 + r) * QPL + t;
    const float* src = qs + (size_t)q * DIM + half * 32;
    float* dst = X + r * 68 + half * 32;
#pragma unroll
    for (int c = 0; c < 32; c += 4)
      *(float4*)(dst + c) = *(const float4*)(src + c);
    __syncthreads();

    gru_step<16>(X, 68, H, 68, lkP, lrkP, lb, m, kh, hreg);
    __syncthreads();
#pragma unroll
    for (int j = 0; j < 4; ++j)
#pragma unroll
      for (int v = 0; v < 8; ++v)
        H[(v + 8 * kh) * 68 + j * 16 + m] = hreg[j][v];
    __syncthreads();
  }
#pragma unroll
  for (int j = 0; j < 4; ++j)
#pragma unroll
    for (int v = 0; v < 8; ++v)
      ls[(size_t)(l0 + v + 8 * kh) * DIM + j * 16 + m] = hreg[j][v];
}

// readout MLP: selu(selu(h@W1+b1)@W2+b2)@W3+b3, 16 paths / wave
__global__ __launch_bounds__(32)
void readout_k(const float* __restrict__ ps, const float* __restrict__ W1p,
               const float* __restrict__ b1, const float* __restrict__ W2p,
               const float* __restrict__ b2, const float* __restrict__ W3,
               const float* __restrict__ b3, float* __restrict__ out) {
  __shared__ float H0[16 * 68];
  __shared__ float H1[16 * 260];
  __shared__ float H2[16 * 260];
  const int lane = threadIdx.x;
  const int m = lane & 15, kh = lane >> 4;
  const int p0 = blockIdx.x * 16;

  const int r = lane >> 1, half = lane & 1;
  {
    const float* src = ps + (size_t)(p0 + r) * DIM + half * 32;
    float* dst = H0 + r * 68 + half * 32;
#pragma unroll
    for (int c = 0; c < 32; c += 4)
      *(float4*)(dst + c) = *(const float4*)(src + c);
  }
  __syncthreads();

#pragma unroll 1
  for (int np = 0; np < 8; ++np) {  // h1 = selu(h0 @ W1 + b1)  [16 x 256]
    v8f c0 = {}, c1 = {};
    gemm_dual<16>(H0, 68, W1p, RU, np * 32, np * 32 + 16, m, kh, c0, c1);
    float ba = b1[np * 32 + m];
    float bb = b1[np * 32 + 16 + m];
#pragma unroll
    for (int v = 0; v < 8; ++v) {
      H1[(v + 8 * kh) * 260 + np * 32 + m] = selu_f(c0[v] + ba);
      H1[(v + 8 * kh) * 260 + np * 32 + 16 + m] = selu_f(c1[v] + bb);
    }
  }
  __syncthreads();

#pragma unroll 1
  for (int np = 0; np < 8; ++np) {  // h2 = selu(h1 @ W2 + b2)  [16 x 256]
    v8f c0 = {}, c1 = {};
    gemm_dual<64>(H1, 260, W2p, RU, np * 32, np * 32 + 16, m, kh, c0, c1);
    float ba = b2[np * 32 + m];
    float bb = b2[np * 32 + 16 + m];
#pragma unroll
    for (int v = 0; v < 8; ++v) {
      H2[(v + 8 * kh) * 260 + np * 32 + m] = selu_f(c0[v] + ba);
      H2[(v + 8 * kh) * 260 + np * 32 + 16 + m] = selu_f(c1[v] + bb);
    }
  }
  __syncthreads();

  if (lane < 16) {  // out = h2 @ W3 + b3  [16 x 1]
    float sum = b3[0];
#pragma unroll 4
    for (int k = 0; k < RU; ++k) sum += H2[m * 260 + k] * W3[k];
    out[p0 + m] = sum;
  }
}

// ---------------- launcher ----------------

extern "C" void kernel_launch(void* const* d_in, const int* in_sizes, int n_in,
                              void* d_out, int out_size, void* d_ws,
                              size_t ws_size, hipStream_t stream) {
  const float* traffic = (const float*)d_in[0];
  const float* packets = (const float*)d_in[1];
  const float* capacity = (const float*)d_in[2];
  const float* qsize = (const float*)d_in[3];
  const float* weight = (const float*)d_in[4];
  const int* policy = (const int*)d_in[5];
  const int* priority = (const int*)d_in[6];
  const int* link_to_path = (const int*)d_in[7];
  const int* queue_to_path = (const int*)d_in[8];
  const int* path_to_queue = (const int*)d_in[12];
  const int* sequence_queues = (const int*)d_in[13];
  const float* pk = (const float*)d_in[17];
  const float* prk = (const float*)d_in[18];
  const float* pb = (const float*)d_in[19];
  const float* qk = (const float*)d_in[20];
  const float* qrk = (const float*)d_in[21];
  const float* qb = (const float*)d_in[22];
  const float* lk = (const float*)d_in[23];
  const float* lrk = (const float*)d_in[24];
  const float* lb = (const float*)d_in[25];
  const float* W1 = (const float*)d_in[26];
  const float* b1 = (const float*)d_in[27];
  const float* W2 = (const float*)d_in[28];
  const float* b2 = (const float*)d_in[29];
  const float* W3 = (const float*)d_in[30];
  const float* b3 = (const float*)d_in[31];
  float* outp = (float*)d_out;

  // workspace carve-up (floats)
  float* W = (float*)d_ws;
  size_t o = 0;
  float* ps = W + o;    o += (size_t)N_P * DIM;
  float* qs = W + o;    o += (size_t)N_QUE * DIM;
  float* ls = W + o;    o += (size_t)N_LNK * DIM;
  float* psum = W + o;  o += (size_t)N_QUE * DIM;
  float* pkP = W + o;   o += (size_t)128 * GDIM;   // 64 pairs x 192 x2
  float* prkP = W + o;  o += (size_t)64 * GDIM;    // 32 pairs x 192 x2
  float* qkP = W + o;   o += (size_t)64 * GDIM;
  float* qrkP = W + o;  o += (size_t)64 * GDIM;
  float* lkP = W + o;   o += (size_t)64 * GDIM;
  float* lrkP = W + o;  o += (size_t)64 * GDIM;
  float* W1p = W + o;   o += (size_t)64 * RU;      // 32 pairs x 256 x2
  float* W2p = W + o;   o += (size_t)256 * RU;     // 128 pairs x 256 x2
  (void)ws_size; (void)n_in; (void)in_sizes; (void)out_size;

  // init states
  init_path_k<<<(N_P * DIM + 255) / 256, 256, 0, stream>>>(traffic, packets, ps);
  init_queue_k<<<(N_QUE * DIM + 255) / 256, 256, 0, stream>>>(qsize, weight,
                                                              priority, qs);
  init_link_k<<<(N_LNK * DIM + 255) / 256, 256, 0, stream>>>(capacity, policy, ls);

  // pack weights into WMMA-friendly pair layout
  pair_weights_k<<<(64 * GDIM + 255) / 256, 256, 0, stream>>>(pk, pkP, 128, GDIM);
  pair_weights_k<<<(32 * GDIM + 255) / 256, 256, 0, stream>>>(prk, prkP, 64, GDIM);
  pair_weights_k<<<(32 * GDIM + 255) / 256, 256, 0, stream>>>(qk, qkP, 64, GDIM);
  pair_weights_k<<<(32 * GDIM + 255) / 256, 256, 0, stream>>>(qrk, qrkP, 64, GDIM);
  pair_weights_k<<<(32 * GDIM + 255) / 256, 256, 0, stream>>>(lk, lkP, 64, GDIM);
  pair_weights_k<<<(32 * GDIM + 255) / 256, 256, 0, stream>>>(lrk, lrkP, 64, GDIM);
  pair_weights_k<<<(32 * RU + 255) / 256, 256, 0, stream>>>(W1, W1p, 64, RU);
  pair_weights_k<<<(128 * RU + 255) / 256, 256, 0, stream>>>(W2, W2p, 256, RU);

  for (int it = 0; it < T_IT; ++it) {
    path_update_k<<<N_P / 16, 32, 0, stream>>>(ps, qs, ls, pkP, prkP, pb,
                                               link_to_path, queue_to_path);
    zero_k<<<(N_QUE * DIM + 255) / 256, 256, 0, stream>>>(psum, N_QUE * DIM);
    scatter_add_k<<<(N_E * DIM + 255) / 256, 256, 0, stream>>>(
        ps, path_to_queue, sequence_queues, psum);
    queue_update_k<<<N_QUE / 16, 32, 0, stream>>>(qs, psum, qkP, qrkP, qb);
    link_update_k<<<N_LNK / 16, 32, 0, stream>>>(ls, qs, lkP, lrkP, lb);
  }

  readout_k<<<N_P / 16, 32, 0, stream>>>(ps, W1p, b1, W2p, b2, W3, b3, outp);
}